// Mosaic_50603304682317
// MI455X (gfx1250) — compile-verified
//
#include <hip/hip_runtime.h>

typedef float v2f __attribute__((ext_vector_type(2)));
typedef float v8f __attribute__((ext_vector_type(8)));

#define BROWS 256   // rows of x
#define IDIM  8192  // inner dim
#define ODIM  8192  // output dim
#define MS    64    // block (tile) size
#define IC    128   // input block count
#define OC    128   // output block count
#define BP    68    // transposed-B LDS pitch (even -> 8B-aligned b64;
                    // banks (4n+ka) cover all 64 -> conflict-free reads)

__global__ __launch_bounds__(128)
void mosaic_wmma_f32(const float* __restrict__ x,
                     const float* __restrict__ blocks,
                     const float* __restrict__ bias,
                     const int*   __restrict__ parts,
                     float*       __restrict__ out) {
  __shared__ float lds_x[MS * MS];    // 16 KB: X tile, row-major (64 rows x 64 k)
  __shared__ float lds_bt[MS * BP];   // 17 KB: weight block TRANSPOSED (n-major)

  const int o    = blockIdx.x;        // output col-block 0..127
  const int mt   = blockIdx.y;        // 64-row tile of x: 0..3
  const int tid  = threadIdx.x;       // 0..127 (4 wave32 waves)
  const int lane = tid & 31;
  const int wave = tid >> 5;

  const int mq       = (wave & 1) * 32;   // wave's 32-row quadrant within 64
  const int nq       = (wave >> 1) * 32;  // wave's 32-col quadrant within 64
  const int lanehalf = lane >> 4;         // 0 or 1
  const int lmod     = lane & 15;

  // 2x2 register blocking: 32x32 output per wave, 4 16x16 accumulators
  v8f acc00 = {};  // rows [mq,mq+16)    x cols [nq,nq+16)
  v8f acc01 = {};  // rows [mq,mq+16)    x cols [nq+16,nq+32)
  v8f acc10 = {};  // rows [mq+16,mq+32) x cols [nq,nq+16)
  v8f acc11 = {};  // rows [mq+16,mq+32) x cols [nq+16,nq+32)

  const float* xband = x + (size_t)(mt * MS) * IDIM;

  for (int i = 0; i < IC; ++i) {
    const int p = parts[i * OC + o];                 // uniform scalar gather index
    const float4* gb = (const float4*)(blocks + (size_t)p * (MS * MS));
    const float*  gx = xband + i * MS;

    // prefetch next gathered block while we work on this one
    {
      const int inext = (i + 1 < IC) ? (i + 1) : i;
      const int pn = parts[inext * OC + o];
      const float* nb = blocks + (size_t)pn * (MS * MS);
      __builtin_prefetch(nb + tid * 32, 0, 3);       // 128 lanes x 128B -> 16 KB block
    }

    // cooperative stage: X verbatim (b128 stores), B transposed (2addr b32 stores)
    #pragma unroll
    for (int c4 = 0; c4 < 8; ++c4) {
      const int l4 = tid + c4 * 128;                 // float4 index 0..1023
      const int r  = l4 >> 4;                        // block row (k dim) 0..63
      const int cc = (l4 & 15) << 2;                 // block col (n dim) 0..60 step 4
      ((float4*)lds_x)[l4] = *(const float4*)(gx + (size_t)r * IDIM + cc);
      const float4 vb = gb[l4];                      // coalesced row-major read
      lds_bt[(cc + 0) * BP + r] = vb.x;              // transpose into n-major
      lds_bt[(cc + 1) * BP + r] = vb.y;
      lds_bt[(cc + 2) * BP + r] = vb.z;
      lds_bt[(cc + 3) * BP + r] = vb.w;
    }
    __syncthreads();

    // 16 k-steps; per step: 2 A + 2 B b64 fragment loads feed 4 WMMAs
    #pragma unroll
    for (int k = 0; k < MS; k += 4) {
      const int ka = k + 2 * lanehalf;               // K = vgpr + 2*lanehalf
      v2f a0 = *(const v2f*)&lds_x [(mq      + lmod) * MS + ka];
      v2f a1 = *(const v2f*)&lds_x [(mq + 16 + lmod) * MS + ka];
      v2f b0 = *(const v2f*)&lds_bt[(nq      + lmod) * BP + ka];
      v2f b1 = *(const v2f*)&lds_bt[(nq + 16 + lmod) * BP + ka];
      acc00 = __builtin_amdgcn_wmma_f32_16x16x4_f32(false, a0, false, b0,
                                                    (short)0, acc00, false, false);
      acc01 = __builtin_amdgcn_wmma_f32_16x16x4_f32(false, a0, false, b1,
                                                    (short)0, acc01, false, false);
      acc10 = __builtin_amdgcn_wmma_f32_16x16x4_f32(false, a1, false, b0,
                                                    (short)0, acc10, false, false);
      acc11 = __builtin_amdgcn_wmma_f32_16x16x4_f32(false, a1, false, b1,
                                                    (short)0, acc11, false, false);
    }
    __syncthreads();
  }

  // Epilogue: C/D layout -> M = vgpr + 8*lanehalf, N = lmod; add bias, store fp32
  const int gcol = o * MS + nq + lmod;
  const float bs0 = bias[gcol];
  const float bs1 = bias[gcol + 16];
  #pragma unroll
  for (int v = 0; v < 8; ++v) {
    const int grow0 = mt * MS + mq + v + 8 * lanehalf;   // rows of acc0x
    const int grow1 = grow0 + 16;                        // rows of acc1x
    out[(size_t)grow0 * ODIM + gcol]      = acc00[v] + bs0;
    out[(size_t)grow0 * ODIM + gcol + 16] = acc01[v] + bs1;
    out[(size_t)grow1 * ODIM + gcol]      = acc10[v] + bs0;
    out[(size_t)grow1 * ODIM + gcol + 16] = acc11[v] + bs1;
  }
}

extern "C" void kernel_launch(void* const* d_in, const int* in_sizes, int n_in,
                              void* d_out, int out_size, void* d_ws, size_t ws_size,
                              hipStream_t stream) {
  const float* x      = (const float*)d_in[0];   // 256*8192 f32
  const float* blocks = (const float*)d_in[1];   // 1024*64*64 f32
  const float* bias   = (const float*)d_in[2];   // 8192 f32
  const int*   parts  = (const int*)d_in[3];     // 128*128 i32
  float* out = (float*)d_out;                    // 256*8192 f32

  dim3 grid(OC, BROWS / MS);   // 128 x 4 = 512 workgroups
  dim3 block(128);             // 4 wave32 waves, one 32x32 quadrant each
  hipLaunchKernelGGL(mosaic_wmma_f32, grid, block, 0, stream,
                     x, blocks, bias, parts, out);
}